// SemanticAwareConv_15917148799637
// MI455X (gfx1250) — compile-verified
//
#include <hip/hip_runtime.h>
#include <hip/hip_bf16.h>

#define N_NODES 50000
#define N_EDGES 800000
#define DIM 128

typedef float v8f __attribute__((ext_vector_type(8)));
typedef float v2f __attribute__((ext_vector_type(2)));

// ---------------------------------------------------------------------------
// Zero-init output [N,128] and segment-sum accumulator [N]
// ---------------------------------------------------------------------------
__global__ __launch_bounds__(256) void init_zero_kernel(float* __restrict__ out,
                                                        float* __restrict__ seg_sum) {
    int i = blockIdx.x * 256 + threadIdx.x;
    if (i < N_NODES * DIM) out[i] = 0.0f;
    if (i < N_NODES) seg_sum[i] = 0.0f;
}

// ---------------------------------------------------------------------------
// y = x @ W^T + b  via V_WMMA_F32_16X16X4_F32 (full fp32, wave32).
// One wave per 16-row block of x; the wave walks K in steps of 4 and keeps
// all 8 output column tiles (8 x v8f accumulators), reusing each A fragment
// across the 8 B tiles.
//
// A (16x4, MxK) per ISA: lane 0-15 -> M=lane, v0=K0, v1=K1;
//                        lane16-31 -> M=lane-16, v0=K2, v1=K3.
// B (4x16, KxN) mirrors with lane%16 = N. Both are contiguous float2 loads
// from row-major x[N,128] and W[out=128,in=128].
// C/D (16x16 f32): vgpr r holds M=r (lanes 0-15, N=lane) / M=r+8 (lanes 16-31).
// ---------------------------------------------------------------------------
__global__ __launch_bounds__(32) void linear_wmma_kernel(const float* __restrict__ x,
                                                         const float* __restrict__ W,
                                                         const float* __restrict__ bias,
                                                         float* __restrict__ y) {
    const int lane = threadIdx.x;        // 0..31, wave32
    const int half = lane >> 4;          // 0 | 1 -> K pair select
    const int l16  = lane & 15;          // M for A-frag, N for B-frag
    const int rowBase = blockIdx.x * 16; // 3125 blocks * 16 = 50000 exactly

    v8f acc[8] = {};                     // 8 column tiles of 16x16 f32

    const float* arow = x + (size_t)(rowBase + l16) * DIM + 2 * half;

    for (int ko = 0; ko < DIM; ko += 4) {
        v2f a;
        a.x = arow[ko + 0];
        a.y = arow[ko + 1];
#pragma unroll
        for (int t = 0; t < 8; ++t) {
            const float* bp = W + (size_t)(t * 16 + l16) * DIM + ko + 2 * half;
            v2f b;
            b.x = bp[0];
            b.y = bp[1];
            acc[t] = __builtin_amdgcn_wmma_f32_16x16x4_f32(
                false, a, false, b, (short)0, acc[t], false, false);
        }
    }

#pragma unroll
    for (int t = 0; t < 8; ++t) {
        const int n = t * 16 + l16;
        const float bv = bias[n];
#pragma unroll
        for (int r = 0; r < 8; ++r) {
            const int m = r + 8 * half;
            y[(size_t)(rowBase + m) * DIM + n] = acc[t][r] + bv;
        }
    }
}

// ---------------------------------------------------------------------------
// inv_norm[n] = 1 / max(||sem[n]||, 1e-8). One wave per node (4 dims/lane).
// ---------------------------------------------------------------------------
__global__ __launch_bounds__(256) void seminorm_kernel(const float* __restrict__ sem,
                                                       float* __restrict__ inv_norm) {
    const int wave = threadIdx.x >> 5;
    const int lane = threadIdx.x & 31;
    const int n = blockIdx.x * 8 + wave;          // 6250 * 8 = 50000 exactly
    const float4 v = *(const float4*)(sem + (size_t)n * DIM + lane * 4);
    float ss = v.x * v.x + v.y * v.y + v.z * v.z + v.w * v.w;
#pragma unroll
    for (int off = 16; off > 0; off >>= 1) ss += __shfl_xor(ss, off, 32);
    if (lane == 0) inv_norm[n] = 1.0f / fmaxf(sqrtf(ss), 1e-8f);
}

// ---------------------------------------------------------------------------
// Per edge: w[e] = exp(cos_sim(src,dst)); seg_sum[dst] += w[e].
// sim in [-1,1] so exp() is unconditionally stable -> no segment-max pass.
// One wave per edge; sem table (25.6MB) is L2-resident on MI455X.
// ---------------------------------------------------------------------------
__global__ __launch_bounds__(256) void edge_weight_kernel(const int* __restrict__ ei,
                                                          const float* __restrict__ sem,
                                                          const float* __restrict__ inv_norm,
                                                          float* __restrict__ w,
                                                          float* __restrict__ seg_sum) {
    const int wave = threadIdx.x >> 5;
    const int lane = threadIdx.x & 31;
    const int e = blockIdx.x * 8 + wave;          // 100000 * 8 = 800000 exactly
    const int src = ei[e];
    const int dst = ei[N_EDGES + e];
    const float4 a = *(const float4*)(sem + (size_t)src * DIM + lane * 4);
    const float4 b = *(const float4*)(sem + (size_t)dst * DIM + lane * 4);
    float d = a.x * b.x + a.y * b.y + a.z * b.z + a.w * b.w;
#pragma unroll
    for (int off = 16; off > 0; off >>= 1) d += __shfl_xor(d, off, 32);
    if (lane == 0) {
        const float sim = d * inv_norm[src] * inv_norm[dst];
        const float wt = expf(sim);
        w[e] = wt;
        atomicAdd(&seg_sum[dst], wt);
    }
}

// ---------------------------------------------------------------------------
// out[dst] += (w[e]/(seg_sum[dst]+1e-16)) * y[src]. One wave per edge,
// 4 dims/lane, fire-and-forget global f32 atomics (L2-resident out table).
// ---------------------------------------------------------------------------
__global__ __launch_bounds__(256) void aggregate_kernel(const int* __restrict__ ei,
                                                        const float* __restrict__ y,
                                                        const float* __restrict__ w,
                                                        const float* __restrict__ seg_sum,
                                                        float* __restrict__ out) {
    const int wave = threadIdx.x >> 5;
    const int lane = threadIdx.x & 31;
    const int e = blockIdx.x * 8 + wave;
    const int src = ei[e];
    const int dst = ei[N_EDGES + e];
    const float attn = w[e] / (seg_sum[dst] + 1e-16f);
    const float4 v = *(const float4*)(y + (size_t)src * DIM + lane * 4);
    float* op = out + (size_t)dst * DIM + lane * 4;
    atomicAdd(op + 0, attn * v.x);
    atomicAdd(op + 1, attn * v.y);
    atomicAdd(op + 2, attn * v.z);
    atomicAdd(op + 3, attn * v.w);
}

// ---------------------------------------------------------------------------
// Inputs (setup_inputs order): x[N*128], edge_index[2*E], semantic_vec[N*128],
// W_src[128*128], b_src[128], W_dst (unused), b_dst (unused).
// Output: out[N*128] f32. Workspace: y(N*128) + w(E) + seg_sum(N) + inv_norm(N)
// = 29.2 MB.
// ---------------------------------------------------------------------------
extern "C" void kernel_launch(void* const* d_in, const int* in_sizes, int n_in,
                              void* d_out, int out_size, void* d_ws, size_t ws_size,
                              hipStream_t stream) {
    const float* x    = (const float*)d_in[0];
    const int*   ei   = (const int*)d_in[1];
    const float* sem  = (const float*)d_in[2];
    const float* Wsrc = (const float*)d_in[3];
    const float* bsrc = (const float*)d_in[4];
    float* out = (float*)d_out;

    float* y        = (float*)d_ws;                       // N*DIM
    float* wbuf     = y + (size_t)N_NODES * DIM;          // E
    float* seg_sum  = wbuf + N_EDGES;                     // N
    float* inv_norm = seg_sum + N_NODES;                  // N

    init_zero_kernel<<<(N_NODES * DIM + 255) / 256, 256, 0, stream>>>(out, seg_sum);
    linear_wmma_kernel<<<N_NODES / 16, 32, 0, stream>>>(x, Wsrc, bsrc, y);
    seminorm_kernel<<<N_NODES / 8, 256, 0, stream>>>(sem, inv_norm);
    edge_weight_kernel<<<N_EDGES / 8, 256, 0, stream>>>(ei, sem, inv_norm, wbuf, seg_sum);
    aggregate_kernel<<<N_EDGES / 8, 256, 0, stream>>>(ei, y, wbuf, seg_sum, out);
}